// MultiHeadPosAtt_79748952752792
// MI455X (gfx1250) — compile-verified
//
#include <hip/hip_runtime.h>
#include <hip/hip_bf16.h>
#include <math.h>

typedef __attribute__((ext_vector_type(2))) float v2f;
typedef __attribute__((ext_vector_type(8))) float v8f;

#define N_SEQ  4096
#define N_HEAD 8
#define HID    256
#define VDIM   32
#define BATCH  2
#define KC     64            // k-chunk width processed per iteration
#define NC     (N_SEQ / KC)  // 64 chunks
#define TQ     64            // query rows per workgroup (4 WMMA M-tiles)

__device__ __forceinline__ float gelu_tanh(float x) {
    // jax.nn.gelu(approximate=True)
    float x3 = x * x * x;
    return 0.5f * x * (1.0f + tanhf(0.7978845608028654f * (x + 0.044715f * x3)));
}

__device__ __forceinline__ float head_scale(float rv) {
    return tanf(0.25f * 3.14159265358979323846f * (1.0f - 1e-7f) * (1.0f + sinf(rv)));
}

// ---------------------------------------------------------------------------
// Kernel 1: value[b,h,k,v] = sum_j x[b,k,j] * weight[h,j,v]
// grid = BATCH*N_SEQ blocks of 256 threads (thread t -> h = t/32, v = t%32)
// ---------------------------------------------------------------------------
__global__ __launch_bounds__(256) void value_kernel(const float* __restrict__ x,
                                                    const float* __restrict__ w,
                                                    float* __restrict__ value) {
    __shared__ float xs[HID];
    const int tid = threadIdx.x;
    const int bn  = blockIdx.x;            // b * N_SEQ + n
    const int b   = bn >> 12;
    const int n   = bn & (N_SEQ - 1);

    xs[tid] = x[(size_t)bn * HID + tid];
    __syncthreads();

    const int h = tid >> 5;
    const int v = tid & 31;
    const float* wp = w + (size_t)h * HID * VDIM + v;
    float acc = 0.0f;
#pragma unroll 8
    for (int j = 0; j < HID; ++j)
        acc = fmaf(xs[j], wp[(size_t)j * VDIM], acc);

    value[(((size_t)b * N_HEAD + h) * N_SEQ + n) * VDIM + v] = acc;
}

// ---------------------------------------------------------------------------
// Kernel 2: per-(h,q) row: 10th-percentile threshold (interp of order stats
// 409/410 of 4096) and row min, both pre-multiplied by the head scale.
// Radix select (4 passes x 8 bits) on u32 bit patterns (all values >= 0).
// One 256-thread block per row.
// ---------------------------------------------------------------------------
__global__ __launch_bounds__(256) void threshold_kernel(const float* __restrict__ m_dist,
                                                        const float* __restrict__ r,
                                                        float* __restrict__ thr,
                                                        float* __restrict__ minv) {
    __shared__ uint32_t keys[N_SEQ];
    __shared__ uint32_t hist[256];
    __shared__ float    red[256];
    __shared__ uint32_t s_bucket;
    __shared__ int      s_k;

    const int tid   = threadIdx.x;
    const int rowid = blockIdx.x;          // h * N_SEQ + q
    const int h     = rowid >> 12;
    const float* src = m_dist + (size_t)rowid * N_SEQ;

    float lmin = 3.402823466e38f;
#pragma unroll
    for (int i = 0; i < N_SEQ / 256; ++i) {
        float v = src[tid + 256 * i];
        lmin = fminf(lmin, v);
        keys[tid + 256 * i] = __float_as_uint(v);
    }
    red[tid] = lmin;
    __syncthreads();
    for (int off = 128; off > 0; off >>= 1) {
        if (tid < off) red[tid] = fminf(red[tid], red[tid + off]);
        __syncthreads();
    }
    const float rowmin = red[0];

    uint32_t sel[2];
    for (int t = 0; t < 2; ++t) {
        uint32_t prefix = 0;
        if (tid == 0) s_k = 409 + t;       // 0-indexed order statistics 409, 410
        __syncthreads();
        for (int pass = 0; pass < 4; ++pass) {
            const int shift = 24 - 8 * pass;
            hist[tid] = 0;
            __syncthreads();
            for (int i = 0; i < N_SEQ / 256; ++i) {
                uint32_t key = keys[tid + 256 * i];
                uint32_t top = (pass == 0) ? 0u : (key >> (32 - 8 * pass));
                if (top == prefix)
                    atomicAdd(&hist[(key >> shift) & 255u], 1u);
            }
            __syncthreads();
            if (tid == 0) {
                int k = s_k;
                int run = 0;
                uint32_t bsel = 255u;
                for (int bkt = 0; bkt < 256; ++bkt) {
                    int c = (int)hist[bkt];
                    if (run + c > k) { bsel = (uint32_t)bkt; s_k = k - run; break; }
                    run += c;
                }
                s_bucket = bsel;
            }
            __syncthreads();
            prefix = (prefix << 8) | s_bucket;
            __syncthreads();
        }
        sel[t] = prefix;
    }

    if (tid == 0) {
        const float scl = head_scale(r[h]);
        const float v0 = __uint_as_float(sel[0]);
        const float v1 = __uint_as_float(sel[1]);
        thr[rowid]  = 0.5f * (v0 + v1) * scl;   // linear interp at 409.5
        minv[rowid] = rowmin * scl;
    }
}

// ---------------------------------------------------------------------------
// Kernel 3: fused masked-softmax + (att @ value) + GELU.
// One 128-thread block (4 waves) per (head, 64-row q-tile).
// Wave w owns (batch = w>>1, n-tile = w&1) and accumulates FOUR 16x16 f32
// tiles (M-tiles m=0..3) via V_WMMA_F32_16X16X4_F32, sharing one B fragment
// per step. The att tile is double-buffered in LDS so next chunk's m_dist
// loads + exp overlap with current chunk's WMMAs (one barrier per chunk).
// ---------------------------------------------------------------------------
__global__ __launch_bounds__(128) void attention_kernel(const float* __restrict__ m_dist,
                                                        const float* __restrict__ value,
                                                        const float* __restrict__ thr,
                                                        const float* __restrict__ minv,
                                                        const float* __restrict__ r,
                                                        float* __restrict__ out) {
    __shared__ float s_att[2][TQ][KC + 4];   // 2 x 64 x 68 f32 = ~34.8 KB
    __shared__ float s_psum[TQ][2];
    __shared__ float s_denom[TQ];

    const int tid = threadIdx.x;
    const int blk = blockIdx.x;
    const int h   = blk >> 6;                    // N_SEQ/TQ = 64 q-tiles per head
    const int q0  = (blk & 63) * TQ;

    const float scl = head_scale(r[h]);

    // phase-1 role: thread covers (row = tid/2, 32 consecutive k per chunk)
    const int row = tid >> 1;                    // 0..63
    const int sub = tid & 1;                     // halves of the 64-wide chunk
    const float rthr = thr[h * N_SEQ + q0 + row];
    const float rmin = minv[h * N_SEQ + q0 + row];
    const float* mrow = m_dist + ((size_t)h * N_SEQ + q0 + row) * N_SEQ + sub * 32;

    // phase-2 role: wave -> (batch, n-tile); lane layout per ISA 7.12.2
    const int lane  = tid & 31;
    const int wv    = tid >> 5;
    const int batch = wv >> 1;
    const int nt    = wv & 1;
    const int klane = (lane >> 4) << 1;          // 0 for lanes 0-15, 2 for 16-31
    const int arow  = lane & 15;                 // A-matrix M index within tile
    const float* vbase = value + ((size_t)batch * N_HEAD + h) * N_SEQ * VDIM
                               + nt * 16 + (lane & 15);

    v8f acc[4] = {{}, {}, {}, {}};
    float sumw = 0.0f;
    float4 m[8];

    // ---- prologue: load + convert chunk 0 into buffer 0 ----
#pragma unroll
    for (int j = 0; j < 8; ++j)
        m[j] = *(const float4*)(mrow + j * 4);
    {
        float* dst = &s_att[0][row][sub * 32];
#pragma unroll
        for (int j = 0; j < 8; ++j) {
            const float4 mv = m[j];
            const float s0 = mv.x * scl, s1 = mv.y * scl, s2 = mv.z * scl, s3 = mv.w * scl;
            const float w0 = (s0 <= rthr) ? __expf(rmin - s0) : 0.0f;
            const float w1 = (s1 <= rthr) ? __expf(rmin - s1) : 0.0f;
            const float w2 = (s2 <= rthr) ? __expf(rmin - s2) : 0.0f;
            const float w3 = (s3 <= rthr) ? __expf(rmin - s3) : 0.0f;
            dst[j * 4 + 0] = w0; dst[j * 4 + 1] = w1;
            dst[j * 4 + 2] = w2; dst[j * 4 + 3] = w3;
            sumw += (w0 + w1) + (w2 + w3);
        }
    }

    int buf = 0;
    for (int kc = 0; kc < NC; ++kc) {
        __syncthreads();                         // att[buf] visible to all waves
        const int kbase = kc * KC;
        const bool more = (kc + 1 < NC);

        // issue next chunk's m_dist loads early (overlap with WMMAs)
        if (more) {
#pragma unroll
            for (int j = 0; j < 8; ++j)
                m[j] = *(const float4*)(mrow + kbase + KC + j * 4);
        }

        // ---- phase 2: 16 WMMA steps x 4 M-tiles on s_att[buf] ----
#pragma unroll
        for (int s = 0; s < KC / 4; ++s) {
            const int ak = 4 * s + klane;
            const float* vp = vbase + (size_t)(kbase + ak) * VDIM;
            v2f bf;
            bf.x = vp[0];
            bf.y = vp[VDIM];
#pragma unroll
            for (int mt = 0; mt < 4; ++mt) {
                v2f a;
                a.x = s_att[buf][mt * 16 + arow][ak];
                a.y = s_att[buf][mt * 16 + arow][ak + 1];
                acc[mt] = __builtin_amdgcn_wmma_f32_16x16x4_f32(
                    false, a, false, bf, (short)0, acc[mt], false, false);
            }
        }

        // ---- phase 1 (pipelined): weights for chunk kc+1 into other buffer ----
        if (more) {
            float* dst = &s_att[buf ^ 1][row][sub * 32];
#pragma unroll
            for (int j = 0; j < 8; ++j) {
                const float4 mv = m[j];
                const float s0 = mv.x * scl, s1 = mv.y * scl, s2 = mv.z * scl, s3 = mv.w * scl;
                const float w0 = (s0 <= rthr) ? __expf(rmin - s0) : 0.0f;
                const float w1 = (s1 <= rthr) ? __expf(rmin - s1) : 0.0f;
                const float w2 = (s2 <= rthr) ? __expf(rmin - s2) : 0.0f;
                const float w3 = (s3 <= rthr) ? __expf(rmin - s3) : 0.0f;
                dst[j * 4 + 0] = w0; dst[j * 4 + 1] = w1;
                dst[j * 4 + 2] = w2; dst[j * 4 + 3] = w3;
                sumw += (w0 + w1) + (w2 + w3);
            }
            buf ^= 1;
        }
    }

    // ---- denominator reduction ----
    s_psum[row][sub] = sumw;
    __syncthreads();
    if (tid < TQ) s_denom[tid] = s_psum[tid][0] + s_psum[tid][1];
    __syncthreads();

    // ---- epilogue: normalize, GELU, store ----
    const int rbase = (lane >> 4) * 8;           // C layout: VGPR i -> M=i / M=8+i
    const int vcol  = nt * 16 + (lane & 15);
#pragma unroll
    for (int mt = 0; mt < 4; ++mt) {
#pragma unroll
        for (int i = 0; i < 8; ++i) {
            const int rr = mt * 16 + rbase + i;
            float o = acc[mt][i] / s_denom[rr];
            o = gelu_tanh(o);
            out[((size_t)batch * N_SEQ + q0 + rr) * HID + h * VDIM + vcol] = o;
        }
    }
}

// ---------------------------------------------------------------------------
extern "C" void kernel_launch(void* const* d_in, const int* in_sizes, int n_in,
                              void* d_out, int out_size, void* d_ws, size_t ws_size,
                              hipStream_t stream) {
    (void)in_sizes; (void)n_in; (void)out_size; (void)ws_size;

    const float* m_dist = (const float*)d_in[0];   // [8, 4096, 4096]
    const float* x      = (const float*)d_in[1];   // [2, 4096, 256]
    const float* r      = (const float*)d_in[2];   // [8]
    const float* weight = (const float*)d_in[3];   // [8, 256, 32]
    float* out = (float*)d_out;                    // [2, 4096, 256]

    float* value = (float*)d_ws;                                   // 8 MB
    float* thr   = value + (size_t)BATCH * N_HEAD * N_SEQ * VDIM;  // 128 KB
    float* minv  = thr + (size_t)N_HEAD * N_SEQ;                   // 128 KB

    value_kernel<<<BATCH * N_SEQ, 256, 0, stream>>>(x, weight, value);
    threshold_kernel<<<N_HEAD * N_SEQ, 256, 0, stream>>>(m_dist, r, thr, minv);
    attention_kernel<<<N_HEAD * (N_SEQ / TQ), 128, 0, stream>>>(m_dist, value, thr, minv, r, out);
}